// PiecewisePolynomial_81587198755415
// MI455X (gfx1250) — compile-verified
//
#include <hip/hip_runtime.h>

// Problem constants (from the reference).
#define BATCH   512
#define IN_F    256
#define OUT_F   256
#define SEGS    16
#define WK      49     // (N-1)*SEGMENTS + 1 coefficients per (o,i)
#define WKP     52     // padded K-stripe per i (multiple of 4, keeps 8B alignment)
#define BTILE   16     // batch rows per block
#define OTILE   16     // outputs per wave
#define WAVES   8      // waves per block
#define ICHUNK  16     // input features staged in LDS per phase
// LDS row stride in dwords: 16*52 = 832 used + 2 pad.
// Even (8B-aligned float2 reads) and 834 % 64 == 2 -> 16 parallel row reads
// land on distinct banks.
#define ROWSTRIDE 834

typedef float v2f __attribute__((ext_vector_type(2)));
typedef float v8f __attribute__((ext_vector_type(8)));

__global__ __launch_bounds__(256)
void PiecewisePolynomial_kan_wmma(const float* __restrict__ x,
                                  const float* __restrict__ w,
                                  float* __restrict__ out) {
  __shared__ float ldsC[BTILE * ROWSTRIDE];

  const int tid  = threadIdx.x;
  const int lane = tid & 31;          // wave32
  const int wave = tid >> 5;
  const int b0   = blockIdx.x * BTILE;
  const int o0   = (blockIdx.y * WAVES + wave) * OTILE;

  const int nlo = lane & 15;          // matrix row/col within half-wave
  const int hi  = lane >> 4;          // 0: K=0,1  1: K=2,3 (f32 A/B layout)

  v8f acc = {0.f, 0.f, 0.f, 0.f, 0.f, 0.f, 0.f, 0.f};

  // Scatter-phase mapping: one thread per (local batch row, local i).
  const int sb = tid >> 4;            // 0..15 batch row in tile
  const int si = tid & 15;            // 0..15 i within chunk

  for (int chunk = 0; chunk < IN_F / ICHUNK; ++chunk) {
    __syncthreads();                  // previous chunk's readers done

    // ---- Phase 1: zero + scatter basis into LDS (build C tile) ----
    for (int idx = tid; idx < BTILE * ROWSTRIDE; idx += 256) ldsC[idx] = 0.0f;
    __syncthreads();
    {
      const int i  = chunk * ICHUNK + si;
      const float xv = x[(b0 + sb) * IN_F + i];
      int s = (int)floorf((xv + 1.0f) * (0.5f * (float)SEGS));
      s = s < 0 ? 0 : (s > SEGS - 1 ? SEGS - 1 : s);
      const float x_min = (float)s * (2.0f / (float)SEGS) - 1.0f;
      const float t = (xv - x_min) * (float)SEGS - 1.0f;   // local coord in [-1,1]
      // Lagrange basis on Chebyshev-Lobatto nodes {-1,-0.5,0.5,1}
      const float pa = t + 1.0f, pb = t + 0.5f, pc = t - 0.5f, pd = t - 1.0f;
      const float L0 = pb * pc * pd * (-2.0f / 3.0f);
      const float L1 = pa * pc * pd * ( 4.0f / 3.0f);
      const float L2 = pa * pb * pd * (-4.0f / 3.0f);
      const float L3 = pa * pb * pc * ( 2.0f / 3.0f);
      float* dst = &ldsC[sb * ROWSTRIDE + si * WKP + 3 * s];
      dst[0] = L0; dst[1] = L1; dst[2] = L2; dst[3] = L3;
    }
    __syncthreads();

    // ---- Phase 2: f32 WMMA GEMM over this chunk ----
    for (int ii = 0; ii < ICHUNK; ++ii) {
      const int i = chunk * ICHUNK + ii;
      // B-fragment source: this lane's output column o0+nlo, feature i.
      const float* wb = w + ((size_t)(o0 + nlo) * IN_F + i) * WK;
      if (ii + 1 < ICHUNK) __builtin_prefetch(wb + WK, 0, 3);  // global_prefetch_b8
      const float* arow = &ldsC[nlo * ROWSTRIDE + ii * WKP];

#pragma unroll
      for (int step = 0; step < 12; ++step) {
        const int kk = step * 4 + hi * 2;     // lanes 0-15: K=kk,kk+1; 16-31: K=kk+2,kk+3
        v2f a;   a.x   = arow[kk]; a.y   = arow[kk + 1];
        v2f bfr; bfr.x = wb[kk];   bfr.y = wb[kk + 1];
        acc = __builtin_amdgcn_wmma_f32_16x16x4_f32(
            /*neg_a=*/false, a, /*neg_b=*/false, bfr,
            /*c_mod=*/(short)0, acc, /*reuse_a=*/false, /*reuse_b=*/false);
      }
      {
        // step 12 covers K = 48..51: only k=48 exists in w (49..51 are pad).
        // Branchless so EXEC stays all-ones for the WMMA.
        const int kk = 48 + hi * 2;
        v2f a; a.x = arow[kk]; a.y = arow[kk + 1];
        const float v48 = wb[48];
        v2f bfr; bfr.x = hi ? 0.0f : v48; bfr.y = 0.0f;
        acc = __builtin_amdgcn_wmma_f32_16x16x4_f32(
            false, a, false, bfr, (short)0, acc, false, false);
      }
    }
  }

  // ---- Store 16x16 f32 D tile: VGPR r -> M=r (lanes 0-15), M=r+8 (lanes 16-31) ----
#pragma unroll
  for (int r = 0; r < 8; ++r) {
    out[(size_t)(b0 + hi * 8 + r) * OUT_F + (o0 + nlo)] = acc[r];
  }
}

extern "C" void kernel_launch(void* const* d_in, const int* in_sizes, int n_in,
                              void* d_out, int out_size, void* d_ws, size_t ws_size,
                              hipStream_t stream) {
  (void)in_sizes; (void)n_in; (void)d_ws; (void)ws_size; (void)out_size;
  const float* x = (const float*)d_in[0];   // (512, 256)
  const float* w = (const float*)d_in[1];   // (256, 256, 49)
  float* out = (float*)d_out;               // (512, 256)

  dim3 grid(BATCH / BTILE, OUT_F / (OTILE * WAVES));  // (32, 2)
  dim3 block(256);
  PiecewisePolynomial_kan_wmma<<<grid, block, 0, stream>>>(x, w, out);
}